// BaselineDNN_8864812499472
// MI455X (gfx1250) — compile-verified
//
#include <hip/hip_runtime.h>
#include <math.h>

// Problem constants (match reference)
#define VOCAB 100000
#define DIM   300
#define BATCH 4096
#define SEQL  200
#define HID   1000
#define HIDP  1024      // padded N for branch-free WMMA loads
#define OUTD  3
#define REPD  (2*DIM)   // 600

typedef __attribute__((ext_vector_type(2))) float v2f;
typedef __attribute__((ext_vector_type(8))) float v8f;

// ---------------------------------------------------------------------------
// Kernel 0: W1 [600,1000] -> W1t [1024,600], zero-padded, LDS-tiled transpose.
// Makes GEMM B-fragment loads identical in shape to A-fragment loads
// (aligned b64 per lane) and removes all N-bound guards from the K-loop.
// ---------------------------------------------------------------------------
__global__ __launch_bounds__(256) void transpose_pad_kernel(
    const float* __restrict__ W1, float* __restrict__ W1t)
{
    __shared__ float tile[32][33];              // +1 to kill bank conflicts
    const int kBase = blockIdx.x * 32;          // K dim (600)
    const int nBase = blockIdx.y * 32;          // N dim (1024)
    const int tx = threadIdx.x;                 // 0..31
    const int ty = threadIdx.y;                 // 0..7

    #pragma unroll
    for (int i = ty; i < 32; i += 8) {
        const int k = kBase + i, n = nBase + tx;
        tile[i][tx] = (k < REPD && n < HID) ? W1[(size_t)k * HID + n] : 0.0f;
    }
    __syncthreads();
    #pragma unroll
    for (int i = ty; i < 32; i += 8) {
        const int n = nBase + i, k = kBase + tx;
        if (k < REPD)
            W1t[(size_t)n * REPD + k] = tile[tx][i];
    }
}

// ---------------------------------------------------------------------------
// Kernel 1: embedding gather + mean/max pooling.
// One block per batch row. 320 threads; lane == embedding dim so each emb row
// (1200 B) is read fully coalesced. Indices staged in LDS; prefetch 8 tokens
// ahead (global_prefetch_b8) to hide dependent-gather latency. Table (120 MB)
// is L2-resident (192 MB L2), so steady-state gather runs at L2 bandwidth.
// ---------------------------------------------------------------------------
__global__ __launch_bounds__(320) void pool_kernel(
    const float* __restrict__ emb, const int* __restrict__ x,
    const int* __restrict__ lengths, float* __restrict__ rep)
{
    __shared__ int sidx[SEQL];
    const int b   = blockIdx.x;
    const int tid = threadIdx.x;
    if (tid < SEQL) sidx[tid] = x[b * SEQL + tid];
    __syncthreads();
    if (tid >= DIM) return;

    const int len = lengths[b];
    float sum = 0.0f;
    float mx  = -INFINITY;
    for (int t = 0; t < SEQL; ++t) {
        if (t + 8 < SEQL)
            __builtin_prefetch(emb + (size_t)sidx[t + 8] * DIM + tid, 0, 0);
        const float v = emb[(size_t)sidx[t] * DIM + tid];
        sum += v;
        if (t < len) mx = fmaxf(mx, v);
    }
    rep[(size_t)b * REPD + tid]       = sum / (float)len;   // mean over ALL L
    rep[(size_t)b * REPD + DIM + tid] = mx;                 // max over first len
}

// ---------------------------------------------------------------------------
// Kernel 2: h = relu(rep @ W1 + b1) via V_WMMA_F32_16X16X4_F32.
// Each wave owns a 16(M) x 64(N) supertile. Branch-free inner loop:
// 5x global_load_b64 (1 A + 4 B, identical access shape thanks to W1t)
// + 4x v_wmma per K-step of 4. 4096 wave-tiles, 4 waves / 128-thread block.
//
// Fragment layouts (ISA 7.12.2, 32-bit, wave32):
//   A 16x4 : lane&15 = M row; lane>=16 -> K base +2; VGPR pair = K, K+1
//   B 4x16 : lane&15 = N col; lane>=16 -> K base +2; VGPR pair = K, K+1
//   C 16x16: VGPR r, lanes 0-15 -> M=r, N=lane; lanes 16-31 -> M=r+8, N=lane-16
// ---------------------------------------------------------------------------
__global__ __launch_bounds__(128) void gemm1_wmma_kernel(
    const float* __restrict__ rep, const float* __restrict__ W1t,
    const float* __restrict__ b1, float* __restrict__ h)
{
    const int wave  = blockIdx.x * (blockDim.x >> 5) + (threadIdx.x >> 5);
    const int lane  = threadIdx.x & 31;
    const int mtile = wave & 255;        // 256 M-tiles of 16
    const int ntile = wave >> 8;         // 16 N-supertiles of 64
    const int row0  = mtile * 16;
    const int col0  = ntile * 64;
    const int laneM = lane & 15;
    const int kHi   = (lane >> 4) << 1;  // 0 for lanes 0-15, 2 for 16-31

    const float* __restrict__ arow  = rep + (size_t)(row0 + laneM) * REPD;
    const float* __restrict__ brow0 = W1t + (size_t)(col0 +  0 + laneM) * REPD;
    const float* __restrict__ brow1 = W1t + (size_t)(col0 + 16 + laneM) * REPD;
    const float* __restrict__ brow2 = W1t + (size_t)(col0 + 32 + laneM) * REPD;
    const float* __restrict__ brow3 = W1t + (size_t)(col0 + 48 + laneM) * REPD;

    v8f acc0 = {}, acc1 = {}, acc2 = {}, acc3 = {};

    for (int k = 0; k < REPD; k += 4) {
        const int kb = k + kHi;
        const v2f a  = *(const v2f*)(arow  + kb);
        const v2f b0 = *(const v2f*)(brow0 + kb);
        const v2f b1v = *(const v2f*)(brow1 + kb);
        const v2f b2v = *(const v2f*)(brow2 + kb);
        const v2f b3v = *(const v2f*)(brow3 + kb);
        acc0 = __builtin_amdgcn_wmma_f32_16x16x4_f32(false, a, false, b0,  (short)0, acc0, false, false);
        acc1 = __builtin_amdgcn_wmma_f32_16x16x4_f32(false, a, false, b1v, (short)0, acc1, false, false);
        acc2 = __builtin_amdgcn_wmma_f32_16x16x4_f32(false, a, false, b2v, (short)0, acc2, false, false);
        acc3 = __builtin_amdgcn_wmma_f32_16x16x4_f32(false, a, false, b3v, (short)0, acc3, false, false);
    }

    // Epilogue: bias + ReLU; only here do we guard the padded columns.
    const int mOff = (lane >> 4) << 3;   // lanes 16-31 hold M = r + 8
    const v8f accs[4] = { acc0, acc1, acc2, acc3 };
    #pragma unroll
    for (int j = 0; j < 4; ++j) {
        const int n = col0 + j * 16 + laneM;
        if (n >= HID) continue;
        const float bias = b1[n];
        #pragma unroll
        for (int r = 0; r < 8; ++r) {
            const int m = row0 + r + mOff;
            const float v = accs[j][r] + bias;
            h[(size_t)m * HID + n] = v > 0.0f ? v : 0.0f;
        }
    }
}

// ---------------------------------------------------------------------------
// Kernel 3: out = h @ W2 + b2  ([4096,1000] x [1000,3]). One block per row.
// ---------------------------------------------------------------------------
__global__ __launch_bounds__(256) void gemm2_kernel(
    const float* __restrict__ h, const float* __restrict__ W2,
    const float* __restrict__ b2, float* __restrict__ out)
{
    __shared__ float red[3][256];
    const int b = blockIdx.x, tid = threadIdx.x;
    float a0 = 0.0f, a1 = 0.0f, a2 = 0.0f;
    const float* __restrict__ hr = h + (size_t)b * HID;
    for (int k = tid; k < HID; k += 256) {
        const float v = hr[k];
        a0 += v * W2[k * 3 + 0];
        a1 += v * W2[k * 3 + 1];
        a2 += v * W2[k * 3 + 2];
    }
    red[0][tid] = a0; red[1][tid] = a1; red[2][tid] = a2;
    __syncthreads();
    for (int s = 128; s > 0; s >>= 1) {
        if (tid < s) {
            red[0][tid] += red[0][tid + s];
            red[1][tid] += red[1][tid + s];
            red[2][tid] += red[2][tid + s];
        }
        __syncthreads();
    }
    if (tid < 3) out[b * 3 + tid] = red[tid][0] + b2[tid];
}

// ---------------------------------------------------------------------------
// Host launcher. Inputs (setup_inputs order):
//   0 emb_table f32[100000,300]  1 W1 f32[600,1000]  2 b1 f32[1000]
//   3 W2 f32[1000,3]             4 b2 f32[3]
//   5 x  i32[4096,200]           6 lengths i32[4096]
// Workspace: W1t f32[1024,600] | rep f32[4096,600] | h f32[4096,1000] (~28.7 MB)
// ---------------------------------------------------------------------------
extern "C" void kernel_launch(void* const* d_in, const int* in_sizes, int n_in,
                              void* d_out, int out_size, void* d_ws, size_t ws_size,
                              hipStream_t stream)
{
    const float* emb  = (const float*)d_in[0];
    const float* W1   = (const float*)d_in[1];
    const float* b1   = (const float*)d_in[2];
    const float* W2   = (const float*)d_in[3];
    const float* b2   = (const float*)d_in[4];
    const int*   x    = (const int*)d_in[5];
    const int*   lens = (const int*)d_in[6];
    float* out = (float*)d_out;

    float* W1t = (float*)d_ws;                        // 1024*600 f32
    float* rep = W1t + (size_t)HIDP * REPD;           // 4096*600 f32
    float* h   = rep + (size_t)BATCH * REPD;          // 4096*1000 f32

    // K tiles: ceil(600/32)=19, N tiles: 1024/32=32
    hipLaunchKernelGGL(transpose_pad_kernel, dim3(19, 32), dim3(32, 8), 0, stream,
                       W1, W1t);

    hipLaunchKernelGGL(pool_kernel, dim3(BATCH), dim3(320), 0, stream,
                       emb, x, lens, rep);

    // 4096 wave-tiles (256 M x 16 N-supertiles), 4 waves per 128-thread block
    hipLaunchKernelGGL(gemm1_wmma_kernel, dim3(1024), dim3(128), 0, stream,
                       rep, W1t, b1, h);

    hipLaunchKernelGGL(gemm2_kernel, dim3(BATCH), dim3(256), 0, stream,
                       h, W2, b2, out);
}